// HistMatching_70875550319072
// MI455X (gfx1250) — compile-verified
//
#include <hip/hip_runtime.h>
#include <stdint.h>

// Problem geometry (from reference): B=8 samples, each C*D*H*W = 1*128*160*128 voxels.
#define N_ELEM   2621440u
#define NSAMP    8
#define TILE     2048
#define NBLK     1280u                 // N_ELEM / TILE (exact)
#define NDIG     256
#define SCAN_LEN (NDIG*NBLK)           // 327680
#define SCAN_T   1024
#define CHUNK    (SCAN_LEN/SCAN_T)     // 320 (exact)
#define SCT_T    64                    // scatter threads per block (2 waves)
#define SCT_E    (TILE/SCT_T)          // 32 elements per thread

typedef float v2f __attribute__((ext_vector_type(2)));
typedef float v8f __attribute__((ext_vector_type(8)));

// ---- monotone float <-> uint key transform (total order == float order) ----
__device__ __forceinline__ uint32_t f2key(float f){
  uint32_t u = __float_as_uint(f);
  return u ^ (uint32_t)(((int32_t)u >> 31) | (int32_t)0x80000000);
}
__device__ __forceinline__ float key2f(uint32_t k){
  uint32_t m = (k & 0x80000000u) ? 0x80000000u : 0xFFFFFFFFu;
  return __uint_as_float(k ^ m);
}

// ---- CDNA5 async global->LDS copy (ASYNCcnt tracked), per CDNA5 ISA ----
__device__ __forceinline__ void async_ld16(const void* g, void* lds){
  uint32_t l = (uint32_t)(uintptr_t)lds;           // low 32 bits of generic LDS ptr = LDS offset
  unsigned long long a = (unsigned long long)(uintptr_t)g;
  asm volatile("global_load_async_to_lds_b128 %0, %1, off" :: "v"(l), "v"(a) : "memory");
}
__device__ __forceinline__ void wait_async0(){
  asm volatile("s_wait_asynccnt 0" ::: "memory");
}

// ---------------- kernels ----------------

__global__ __launch_bounds__(256) void k_init(const float* __restrict__ src,
                                              uint32_t* __restrict__ keys,
                                              uint32_t* __restrict__ idxs){
  uint32_t i = blockIdx.x*256u + threadIdx.x;
  keys[i] = f2key(src[i]);
  idxs[i] = i;
}

__global__ __launch_bounds__(256) void k_hist(const uint32_t* __restrict__ keys,
                                              uint32_t* __restrict__ counts, int shift){
  __shared__ uint32_t h[NDIG];
  int t = threadIdx.x, blk = blockIdx.x;
  h[t] = 0;
  __syncthreads();
  const uint32_t* p = keys + (size_t)blk*TILE;
  #pragma unroll
  for (int i = 0; i < TILE/256; ++i){
    uint32_t d = (p[i*256 + t] >> shift) & 255u;
    atomicAdd(&h[d], 1u);
  }
  __syncthreads();
  counts[(size_t)t*NBLK + blk] = h[t];   // digit-major layout
}

// Exclusive scan of counts[SCAN_LEN], single block. Middle phase (scan of the
// 1024 thread-partials) is done with exact f32 WMMA triangular products:
// row-inclusive scan of a 16x16 tile = sum_g A_g(16x4) * B_g(4x16), B_g[k][n]=[n>=4g+k].
__global__ __launch_bounds__(SCAN_T) void k_scan(uint32_t* __restrict__ counts){
  __shared__ uint32_t partial[SCAN_T];
  __shared__ float    dmat[4*256];
  __shared__ float    scanned[SCAN_T];
  __shared__ float    segtotF[4];
  __shared__ uint32_t segoff[4];

  int t = threadIdx.x;
  uint32_t mySum = 0;
  int base = t*CHUNK;
  for (int j = 0; j < CHUNK; ++j) mySum += counts[base + j];
  partial[t] = mySum;
  __syncthreads();

  int wave = t >> 5;        // wave32
  int lane = t & 31;
  if (wave < 4){
    int m  = lane & 15;     // row index (A) / column index (B,D)
    int kh = lane >> 4;     // K-half selector
    const int segBase = wave*256;
    v8f acc = {0.f,0.f,0.f,0.f,0.f,0.f,0.f,0.f};
    #pragma unroll
    for (int g = 0; g < 4; ++g){
      int k0 = 4*g + 2*kh;
      v2f a, b;
      a.x = (float)partial[segBase + m*16 + k0    ];
      a.y = (float)partial[segBase + m*16 + k0 + 1];
      b.x = (m >= k0    ) ? 1.f : 0.f;
      b.y = (m >= k0 + 1) ? 1.f : 0.f;
      acc = __builtin_amdgcn_wmma_f32_16x16x4_f32(false, a, false, b,
                                                  (short)0, acc, false, false);
    }
    // acc[q] = row-inclusive scan, row = q + 8*kh, col = m
    #pragma unroll
    for (int q = 0; q < 8; ++q){
      int row = q + (kh ? 8 : 0);
      dmat[wave*256 + row*16 + m] = acc[q];
    }
    __builtin_amdgcn_wave_barrier();   // LDS is in-order per wave; fence scheduler
    float rt[15];
    #pragma unroll
    for (int r = 0; r < 15; ++r) rt[r] = dmat[wave*256 + r*16 + 15];
    float off[16]; off[0] = 0.f;
    #pragma unroll
    for (int r = 1; r < 16; ++r) off[r] = off[r-1] + rt[r-1];
    #pragma unroll
    for (int q = 0; q < 8; ++q){
      int row = q + (kh ? 8 : 0);
      float v = acc[q] + off[row];
      scanned[segBase + row*16 + m] = v;
      if (row == 15 && m == 15) segtotF[wave] = v;
    }
  }
  __syncthreads();
  if (t == 0){
    uint32_t s = 0;
    for (int w = 0; w < 4; ++w){ segoff[w] = s; s += (uint32_t)segtotF[w]; }
  }
  __syncthreads();
  uint32_t run = (uint32_t)scanned[t] + segoff[t >> 8] - mySum;  // exclusive start
  for (int j = 0; j < CHUNK; ++j){
    uint32_t v = counts[base + j];
    counts[base + j] = run;
    run += v;
  }
}

// Stable per-pass scatter. Tile staged via async global->LDS (b128).
__global__ __launch_bounds__(SCT_T) void k_scatter(const uint32_t* __restrict__ keysIn,
                                                   const uint32_t* __restrict__ idxIn,
                                                   uint32_t* __restrict__ keysOut,
                                                   uint32_t* __restrict__ idxOut,
                                                   const uint32_t* __restrict__ counts,
                                                   int shift){
  __shared__ uint16_t cnt[SCT_T*NDIG];   // 32 KB: per-thread digit counts / bases
  __shared__ uint32_t keybuf[TILE];      // 8 KB
  __shared__ uint32_t idxbuf[TILE];      // 8 KB
  __shared__ uint32_t gbase[NDIG];       // 1 KB

  int t = threadIdx.x, blk = blockIdx.x;
  size_t tb = (size_t)blk*TILE;

  // stage tile into LDS with async tensor-path copies
  #pragma unroll
  for (int i = 0; i < 8; ++i){
    int o = (i*SCT_T + t)*4;
    async_ld16(keysIn + tb + o, &keybuf[o]);
    async_ld16(idxIn  + tb + o, &idxbuf[o]);
  }
  for (int i = t; i < SCT_T*NDIG; i += SCT_T) cnt[i] = 0;
  for (int d = t; d < NDIG; d += SCT_T) gbase[d] = counts[(size_t)d*NBLK + blk];
  wait_async0();
  __syncthreads();

  int myBase = t*SCT_E;
  #pragma unroll 4
  for (int e = 0; e < SCT_E; ++e){
    uint32_t d = (keybuf[myBase + e] >> shift) & 255u;
    cnt[t*NDIG + d]++;                       // slot owned by this thread
  }
  __syncthreads();

  // exclusive column scan over threads for each digit
  for (int d = t; d < NDIG; d += SCT_T){
    uint32_t run = 0;
    for (int tt = 0; tt < SCT_T; ++tt){
      uint32_t c = cnt[tt*NDIG + d];
      cnt[tt*NDIG + d] = (uint16_t)run;
      run += c;
    }
  }
  __syncthreads();

  // stable write-out: in-order within thread, threads ordered by column scan
  for (int e = 0; e < SCT_E; ++e){
    uint32_t k = keybuf[myBase + e];
    uint32_t d = (k >> shift) & 255u;
    uint32_t pos = gbase[d] + (uint32_t)cnt[t*NDIG + d];
    cnt[t*NDIG + d]++;
    keysOut[pos] = k;
    idxOut[pos]  = idxbuf[myBase + e];
  }
}

__global__ __launch_bounds__(256) void k_tmplfin(const uint32_t* __restrict__ keys,
                                                 float* __restrict__ tmpl){
  uint32_t i = blockIdx.x*256u + threadIdx.x;
  tmpl[i] = key2f(keys[i]);
}

__global__ __launch_bounds__(256) void k_write(const uint32_t* __restrict__ idx,
                                               const float* __restrict__ tmpl,
                                               float* __restrict__ out){
  uint32_t r = blockIdx.x*256u + threadIdx.x;
  out[idx[r]] = tmpl[r];      // out[i] = t_sorted[rank(i)]
}

// ---------------- host ----------------
extern "C" void kernel_launch(void* const* d_in, const int* in_sizes, int n_in,
                              void* d_out, int out_size, void* d_ws, size_t ws_size,
                              hipStream_t stream){
  (void)in_sizes; (void)n_in; (void)out_size; (void)ws_size;
  const float* x    = (const float*)d_in[0];
  const float* base = (const float*)d_in[1];
  float* out = (float*)d_out;

  char* ws = (char*)d_ws; size_t off = 0;
  auto carve = [&](size_t bytes)->void*{
    off = (off + 255) & ~(size_t)255;
    void* p = ws + off; off += bytes; return p;
  };
  uint32_t* keyA   = (uint32_t*)carve((size_t)N_ELEM*4);
  uint32_t* idxA   = (uint32_t*)carve((size_t)N_ELEM*4);
  uint32_t* keyB   = (uint32_t*)carve((size_t)N_ELEM*4);
  uint32_t* idxB   = (uint32_t*)carve((size_t)N_ELEM*4);
  uint32_t* counts = (uint32_t*)carve((size_t)SCAN_LEN*4);
  float*    tmpl   = (float*)carve((size_t)N_ELEM*4);

  auto radix_sort = [&](uint32_t*& ka, uint32_t*& ia, uint32_t*& kb, uint32_t*& ib){
    for (int p = 0; p < 4; ++p){
      k_hist   <<<NBLK, 256,   0, stream>>>(ka, counts, p*8);
      k_scan   <<<1,    SCAN_T,0, stream>>>(counts);
      k_scatter<<<NBLK, SCT_T, 0, stream>>>(ka, ia, kb, ib, counts, p*8);
      uint32_t* tk = ka; ka = kb; kb = tk;
      uint32_t* ti = ia; ia = ib; ib = ti;
    }
  };

  { // template: sort once, recover sorted float values
    uint32_t *ka=keyA, *ia=idxA, *kb=keyB, *ib=idxB;
    k_init<<<N_ELEM/256, 256, 0, stream>>>(base, ka, ia);
    radix_sort(ka, ia, kb, ib);
    k_tmplfin<<<N_ELEM/256, 256, 0, stream>>>(ka, tmpl);
  }
  for (int b = 0; b < NSAMP; ++b){
    uint32_t *ka=keyA, *ia=idxA, *kb=keyB, *ib=idxB;
    k_init<<<N_ELEM/256, 256, 0, stream>>>(x + (size_t)b*N_ELEM, ka, ia);
    radix_sort(ka, ia, kb, ib);
    k_write<<<N_ELEM/256, 256, 0, stream>>>(ia, tmpl, out + (size_t)b*N_ELEM);
  }
}